// WongWangDecisionBP_88227218195047
// MI455X (gfx1250) — compile-verified
//
#include <hip/hip_runtime.h>
#include <hip/hip_bf16.h>
#include <stdint.h>

// Wong-Wang decision model, MI455X (gfx1250).
//
// Roofline: mandatory traffic = eps_1+eps_2 streamed once = 164 MB (~7us @ 23.3 TB/s).
// Compute ~0.6 GFLOP. The true bottleneck is the per-element serial 5000-step
// dependency chain (2 transcendentals/step) with only 4096 threads. Strategy:
//  - 1 thread / batch element; s1/s2 chains stay in registers (2-way ILP).
//  - CDNA5 async global->LDS DMA (ASYNCcnt) double-buffers the per-step noise
//    row: step t+2's noise is in flight while step t computes, so memory
//    latency never joins the critical path and no VGPRs hold prefetched data.
//  - Early exit: the signed decision time is fully determined at the FIRST
//    step where either population crosses (later crossings cannot flip
//    t1 < t2), so a wave breaks as soon as all its lanes have decided.
//  - Trip counts templated (5000/4000) so loop branches fold at compile time.

#define WW_BLOCK 256
#define WW_INT_MAX 0x7FFFFFFF

typedef __attribute__((address_space(3))) float lds_f32;

__device__ __forceinline__ uint32_t lds_off_of(float* p) {
  // ptrtoint of an LDS (addrspace(3)) pointer = 32-bit LDS byte offset,
  // exactly what global_load_async_to_lds_b32 wants in its VDST VGPR.
  return (uint32_t)(uintptr_t)(lds_f32*)p;
}

// each lane copies 4 bytes from global address gbase + goff to its LDS byte offset lds_off  (ASYNCcnt++, completes in order)
__device__ __forceinline__ void async_g2l(uint32_t lds_off, uint32_t goff,
                                          const float* gbase) {
  asm volatile("global_load_async_to_lds_b32 %0, %1, %2"
               :
               : "v"(lds_off), "v"(goff), "s"(gbase)
               : "memory");
}

// Same, but with dummy value deps so the overwrite of a double-buffer slot is
// issued only after this iteration's ds_load results are live in registers.
__device__ __forceinline__ void async_g2l_dep(uint32_t lds_off, uint32_t goff,
                                              const float* gbase,
                                              float dep0, float dep1) {
  asm volatile("global_load_async_to_lds_b32 %0, %1, %2"
               :
               : "v"(lds_off), "v"(goff), "s"(gbase), "v"(dep0), "v"(dep1)
               : "memory");
}

template <int CT_TSTEPS, int CT_TSTIM>
__global__ __launch_bounds__(WW_BLOCK)
void WongWang_sim_kernel(const float* __restrict__ sig,
                         const float* __restrict__ pJ11, const float* __restrict__ pJ12,
                         const float* __restrict__ pJ21, const float* __restrict__ pJ22,
                         const float* __restrict__ pJext, const float* __restrict__ pI0,
                         const float* __restrict__ pNoise, const float* __restrict__ pThr,
                         const float* __restrict__ pMdel,
                         const float* __restrict__ eps0_1, const float* __restrict__ eps0_2,
                         const float* __restrict__ eps1, const float* __restrict__ eps2,
                         float* __restrict__ out,
                         int batch, int rtsteps, int rtstim) {
  const int tsteps = (CT_TSTEPS > 0) ? CT_TSTEPS : rtsteps;
  const int tstim  = (CT_TSTEPS > 0) ? CT_TSTIM  : rtstim;

  // layout: [buf0:e1 | buf0:e2 | buf1:e1 | buf1:e2], 256 floats each
  __shared__ float smem[4 * WW_BLOCK];

  const int tid = threadIdx.x;
  const int b   = blockIdx.x * WW_BLOCK + tid;
  if (b >= batch) return;

  // uniform scalar params (s_load)
  const float J11 = *pJ11, J12 = *pJ12, J21 = *pJ21, J22 = *pJ22;
  const float Jext = *pJext, I0 = *pI0, noise = *pNoise;
  const float thr = *pThr, mdel = *pMdel;

  const float sig0 = sig[b];
  const float I1s = Jext * (1.0f + sig0 * 0.01f);   // stimulus currents
  const float I2s = Jext * (1.0f - sig0 * 0.01f);

  const float decay = __expf(-0.25f);                               // exp(-DT/TAU_AMPA)
  const float kn    = noise * sqrtf((1.0f - __expf(-0.5f)) * 0.5f); // noise_ampa*nscale

  float s1 = 0.1f, s2 = 0.1f;
  float In1 = eps0_1[b] * noise;
  float In2 = eps0_2[b] * noise;

  // LDS byte offsets for this lane's four slots
  const uint32_t l0    = lds_off_of(&smem[tid]);
  const uint32_t le1b0 = l0;                 // buf0 e1
  const uint32_t le2b0 = l0 + 1024u;         // buf0 e2
  const uint32_t le1b1 = l0 + 2048u;         // buf1 e1
  const uint32_t le2b1 = l0 + 3072u;         // buf1 e2

  const uint32_t rowbytes = (uint32_t)batch * 4u;   // stride between timesteps
  const uint32_t goff0    = (uint32_t)b * 4u;

  // ---- prologue: launch async copies for t=0 (buf0) and t=1 (buf1) ----
  async_g2l(le1b0, goff0, eps1);
  async_g2l(le2b0, goff0, eps2);
  if (tsteps > 1) {
    async_g2l(le1b1, goff0 + rowbytes, eps1);
    async_g2l(le2b1, goff0 + rowbytes, eps2);
  }
  uint32_t goff_pf = goff0 + 2u * rowbytes;   // next row to prefetch (t+2)

  int t1i = WW_INT_MAX;
  int t2i = WW_INT_MAX;

  for (int t = 0; t < tsteps; ++t) {
    const int buf = t & 1;

    // In-order ASYNCcnt completion: <=2 outstanding means the current
    // buffer's pair has landed (the remaining 2 are next buffer's).
    if (t + 1 < tsteps) {
      asm volatile("s_wait_asynccnt 0x2" ::: "memory");
    } else {
      asm volatile("s_wait_asynccnt 0x0" ::: "memory");
    }

    const float e1 = smem[buf * 512 + tid];          // merged ds_load_2addr
    const float e2 = smem[buf * 512 + 256 + tid];

    // Wong-Wang step
    const bool stim = (t < tstim);
    const float x1 = fmaf(J11, s1, fmaf(-J12, s2, I0)) + (stim ? I1s : 0.0f) + In1;
    const float x2 = fmaf(J22, s2, fmaf(-J21, s1, I0)) + (stim ? I2s : 0.0f) + In2;
    const float u1 = fmaf(270.0f, x1, -108.0f);
    const float u2 = fmaf(270.0f, x2, -108.0f);
    const float d1 = 1.0f - __expf(-0.154f * u1) + 1e-6f;
    const float d2 = 1.0f - __expf(-0.154f * u2) + 1e-6f;
    const float h1 = fmaxf(u1 / d1, 0.0f);
    const float h2 = fmaxf(u2 / d2, 0.0f);
    const float ds1 = fmaf(1.0f - s1, h1 * 0.000641f, -s1 * 0.01f); // gamma/1000, 1/tau_s
    const float ds2 = fmaf(1.0f - s2, h2 * 0.000641f, -s2 * 0.01f);
    In1 = fmaf(kn, e1, In1 * decay);
    In2 = fmaf(kn, e2, In2 * decay);
    s1 = fmaf(ds1, 0.5f, s1);   // * DT
    s2 = fmaf(ds2, 0.5f, s2);

    // branchless first-crossing (post-update state, matching traj rows)
    t1i = (s1 > thr) ? (t < t1i ? t : t1i) : t1i;
    t2i = (s2 > thr) ? (t < t2i ? t : t2i) : t2i;

    // Early exit: outcome is fixed once either population has crossed
    // (a later t1 can only be > t2, leaving "t1 < t2" unchanged).
    const bool undecided = (t1i == WW_INT_MAX) & (t2i == WW_INT_MAX);
    if (__builtin_amdgcn_ballot_w32(undecided) == 0u) break;

    // refill this buffer for step t+2 (deps order it after e1/e2 consumption)
    if (t + 2 < tsteps) {
      const uint32_t le1 = buf ? le1b1 : le1b0;
      const uint32_t le2 = buf ? le2b1 : le2b0;
      async_g2l_dep(le1, goff_pf, eps1, e1, e2);
      async_g2l_dep(le2, goff_pf, eps2, e1, e2);
      goff_pf += rowbytes;
    }
  }
  // NB: on break, outstanding async LDS writes target only this block's LDS
  // and are drained by s_endpgm's implicit wait-idle — safe to leave pending.

  // decision: dtimes = t1 < t2 ? t1 : -t2 (inf if never crossed), ms = *DT,
  // then seconds +/- motor delay by sign.
  const float ft1 = (t1i == WW_INT_MAX) ? __builtin_inff() : (float)t1i;
  const float ft2 = (t2i == WW_INT_MAX) ? __builtin_inff() : (float)t2i;
  const float dtimes = (ft1 < ft2) ? ft1 : -ft2;
  const float ms = dtimes * 0.5f;   // DT
  float res;
  if (ms < 0.0f)      res = ms * 1e-3f - mdel;
  else if (ms > 0.0f) res = ms * 1e-3f + mdel;
  else                res = ms;     // crossing at step 0 (ms = +/-0)
  out[b] = res;
}

extern "C" void kernel_launch(void* const* d_in, const int* in_sizes, int n_in,
                              void* d_out, int out_size, void* d_ws, size_t ws_size,
                              hipStream_t stream) {
  const float* sig    = (const float*)d_in[0];
  const float* J11    = (const float*)d_in[1];
  const float* J12    = (const float*)d_in[2];
  const float* J21    = (const float*)d_in[3];
  const float* J22    = (const float*)d_in[4];
  const float* Jext   = (const float*)d_in[5];
  const float* I0     = (const float*)d_in[6];
  const float* noise  = (const float*)d_in[7];
  const float* thr    = (const float*)d_in[8];
  const float* mdel   = (const float*)d_in[9];
  const float* eps0_1 = (const float*)d_in[10];
  const float* eps0_2 = (const float*)d_in[11];
  const float* eps1   = (const float*)d_in[12];
  const float* eps2   = (const float*)d_in[13];
  float* out = (float*)d_out;

  const int batch  = in_sizes[0];               // 4096
  const int tsteps = in_sizes[12] / batch;      // 5000
  const int tstim  = 4000;                      // int(2000/DT), fixed in reference

  const int grid = (batch + WW_BLOCK - 1) / WW_BLOCK;
  if (tsteps == 5000) {
    WongWang_sim_kernel<5000, 4000><<<grid, WW_BLOCK, 0, stream>>>(
        sig, J11, J12, J21, J22, Jext, I0, noise, thr, mdel,
        eps0_1, eps0_2, eps1, eps2, out, batch, tsteps, tstim);
  } else {
    WongWang_sim_kernel<0, 0><<<grid, WW_BLOCK, 0, stream>>>(
        sig, J11, J12, J21, J22, Jext, I0, noise, thr, mdel,
        eps0_1, eps0_2, eps1, eps2, out, batch, tsteps, tstim);
  }
}